// LocalSpatialEncoding_15582141350273
// MI455X (gfx1250) — compile-verified
//
#include <hip/hip_runtime.h>

// LocalSpatialEncoding for MI455X (gfx1250, wave32).
// B=4, N=65536, K=16, d=8, dim_in=10, dim_out=8.
// Pass 1: deterministic per-block BN partial sums (scalar, SGPR-uniform weights).
// Pass 1.5: finalize mean / (invstd*gamma) / beta into 24-float stats block.
// Pass 2: one wave per (b,n); 16 neighbors -> one 16x16 tile; conv done with
//         three chained V_WMMA_F32_16X16X4_F32 (K padded 10->12), BN+LeakyReLU
//         epilogue, gathers + 16-wide contiguous stores.

#define NPTS   65536
#define BK     16
#define NB     4
#define TOTAL  (NB * NPTS * BK)   // 4194304
#define P1_BLOCKS  1024
#define P1_THREADS 256
#define EPSV   1e-6f
#define SLOPE  0.2f

typedef __attribute__((ext_vector_type(2))) float v2f;
typedef __attribute__((ext_vector_type(8))) float v8f;

// ---------------- Pass 1: BN partial sums ----------------
__global__ void __launch_bounds__(P1_THREADS)
lse_bn_partial(const float* __restrict__ coords,
               const int*   __restrict__ nidx,
               const float* __restrict__ w,
               const float* __restrict__ bias,
               float*       __restrict__ partial) {
    __shared__ float sm[P1_THREADS * 16];
    float sum[8], sq[8];
#pragma unroll
    for (int c = 0; c < 8; ++c) { sum[c] = 0.f; sq[c] = 0.f; }

    const int stride = gridDim.x * blockDim.x;
    for (int e = blockIdx.x * blockDim.x + threadIdx.x; e < TOTAL; e += stride) {
        const int b   = e >> 20;                 // N*K = 2^20
        const int rem = e & (NPTS * BK - 1);
        const int n   = rem >> 4;
        const int k   = rem & 15;
        const int base = b * NPTS;
        const int j = nidx[(base + n) * BK + k];
        const float* cc = coords + (size_t)(base + n) * 3;
        const float* nc = coords + (size_t)(base + j) * 3;
        const float cx = cc[0], cy = cc[1], cz = cc[2];
        const float nx = nc[0], ny = nc[1], nz = nc[2];
        const float rx = cx - nx, ry = cy - ny, rz = cz - nz;
        const float dist = sqrtf(rx * rx + ry * ry + rz * rz);
        const float f[10] = {dist, rx, ry, rz, cx, cy, cz, nx, ny, nz};
#pragma unroll
        for (int c = 0; c < 8; ++c) {
            float x = bias[c];
#pragma unroll
            for (int i = 0; i < 10; ++i) x += w[c * 10 + i] * f[i];
            sum[c] += x;
            sq[c]  += x * x;
        }
    }
#pragma unroll
    for (int c = 0; c < 8; ++c) {
        sm[threadIdx.x * 16 + c]     = sum[c];
        sm[threadIdx.x * 16 + 8 + c] = sq[c];
    }
    __syncthreads();
    const int t = threadIdx.x;
    if (t < 16) {
        float a = 0.f;
        for (int j = 0; j < P1_THREADS; ++j) a += sm[j * 16 + t];
        partial[blockIdx.x * 16 + t] = a;
    }
}

// ---------------- Pass 1.5: finalize BN stats ----------------
__global__ void lse_bn_finalize(const float* __restrict__ partial,
                                const float* __restrict__ gamma,
                                const float* __restrict__ beta,
                                float*       __restrict__ stats) {
    const int c = threadIdx.x;
    if (c < 8) {
        float s = 0.f, q = 0.f;
        for (int j = 0; j < P1_BLOCKS; ++j) {
            s += partial[j * 16 + c];
            q += partial[j * 16 + 8 + c];
        }
        const float inv  = 1.0f / (float)TOTAL;
        const float mean = s * inv;
        const float var  = q * inv - mean * mean;
        const float istd = rsqrtf(var + EPSV);
        stats[c]      = mean;
        stats[8 + c]  = istd * gamma[c];
        stats[16 + c] = beta[c];
    }
}

// ---------------- Pass 2: WMMA conv + epilogue + gathers ----------------
__global__ void __launch_bounds__(256)
lse_main(const float* __restrict__ coords,
         const float* __restrict__ features,
         const int*   __restrict__ nidx,
         const float* __restrict__ w,
         const float* __restrict__ bias,
         const float* __restrict__ stats,
         float*       __restrict__ out) {
    const int lane = threadIdx.x & 31;
    const int p    = lane & 15;   // neighbor index k / matrix column N
    const int hi   = lane >> 4;   // wave half selects K sub-rows {0,1} vs {2,3}
    const int wave = (blockIdx.x * blockDim.x + threadIdx.x) >> 5;
    const int b    = wave >> 16;          // N = 65536
    const int n    = wave & (NPTS - 1);

    const int base = b * NPTS;
    const int j = nidx[(base + n) * BK + p];
    const float* cc = coords + (size_t)(base + n) * 3;
    const float* nc = coords + (size_t)(base + j) * 3;
    const float cx = cc[0], cy = cc[1], cz = cc[2];
    const float nx = nc[0], ny = nc[1], nz = nc[2];
    const float rx = cx - nx, ry = cy - ny, rz = cz - nz;
    const float dist = sqrtf(rx * rx + ry * ry + rz * rz);

    // B operand (12x16, K padded): lane = 16*(K>=2)+N, vgpr = K%2.
    // feature order: {dist, rx, ry, rz, cx, cy, cz, nx, ny, nz, 0, 0}
    v2f b0, b1, b2;
    b0.x = hi ? ry  : dist;  b0.y = hi ? rz  : rx;   // K = 0..3
    b1.x = hi ? cz  : cx;    b1.y = hi ? nx  : cy;   // K = 4..7
    b2.x = hi ? 0.f : ny;    b2.y = hi ? 0.f : nz;   // K = 8..11

    // A operand (16x12, M padded 8->16): lane = 16*(K>=2)+M, vgpr = K%2.
    const int kb = 2 * hi;
    auto wl = [&](int kk) -> float {
        return (p < 8 && kk < 10) ? w[p * 10 + kk] : 0.f;
    };
    v2f a0, a1, a2;
    a0.x = wl(kb);     a0.y = wl(kb + 1);
    a1.x = wl(4 + kb); a1.y = wl(4 + kb + 1);
    a2.x = wl(8 + kb); a2.y = wl(8 + kb + 1);

    // C init = bias: D layout has channel r in VGPR r for lanes 0-15.
    v8f c;
#pragma unroll
    for (int r = 0; r < 8; ++r) c[r] = hi ? 0.f : bias[r];

    // D = A x B + C over K=12 in three K=4 steps (EXEC is all-1s here).
    c = __builtin_amdgcn_wmma_f32_16x16x4_f32(false, a0, false, b0, (short)0, c, false, false);
    c = __builtin_amdgcn_wmma_f32_16x16x4_f32(false, a1, false, b1, (short)0, c, false, false);
    c = __builtin_amdgcn_wmma_f32_16x16x4_f32(false, a2, false, b2, (short)0, c, false, false);

    if (!hi) {
        const size_t NK  = (size_t)NPTS * BK;            // 1048576
        const size_t pos = (size_t)n * BK + p;
        float* out2 = out + (size_t)NB * 16 * NK;        // rel_out tuple element

        // neighbor feature gather -> channels 0..7 of concat output (L2-resident)
#pragma unroll
        for (int ch = 0; ch < 8; ++ch) {
            const float v = features[(size_t)(b * 8 + ch) * NPTS + j];
            out[(size_t)(b * 16 + ch) * NK + pos] = v;
        }
        // BN + LeakyReLU epilogue; stats are uniform per accumulator register
#pragma unroll
        for (int r = 0; r < 8; ++r) {
            const float mean  = stats[r];
            const float scale = stats[8 + r];
            const float shift = stats[16 + r];
            float y = (c[r] - mean) * scale + shift;
            y = y > 0.f ? y : SLOPE * y;
            out[(size_t)(b * 16 + 8 + r) * NK + pos] = y;   // channels 8..15
            out2[(size_t)(b * 8 + r) * NK + pos]     = y;   // rel_out
        }
    }
}

extern "C" void kernel_launch(void* const* d_in, const int* in_sizes, int n_in,
                              void* d_out, int out_size, void* d_ws, size_t ws_size,
                              hipStream_t stream) {
    const float* coords   = (const float*)d_in[0];
    const float* features = (const float*)d_in[1];
    const int*   nidx     = (const int*)d_in[2];
    const float* w        = (const float*)d_in[3];
    const float* bias     = (const float*)d_in[4];
    const float* gamma    = (const float*)d_in[5];
    const float* beta     = (const float*)d_in[6];
    float* out = (float*)d_out;

    float* partial = (float*)d_ws;                 // 1024*16 floats = 64 KB
    float* stats   = partial + P1_BLOCKS * 16;     // 24 floats

    lse_bn_partial<<<P1_BLOCKS, P1_THREADS, 0, stream>>>(coords, nidx, w, bias, partial);
    lse_bn_finalize<<<1, 32, 0, stream>>>(partial, gamma, beta, stats);

    const int waves  = NB * NPTS;                  // 262144 waves, one per (b,n)
    const int thr    = 256;                        // 8 waves per block
    const int blocks = (waves * 32) / thr;         // 32768
    lse_main<<<blocks, thr, 0, stream>>>(coords, features, nidx, w, bias, stats, out);
}